// AAF_Loss_90890097918293
// MI455X (gfx1250) — compile-verified
//
#include <hip/hip_runtime.h>
#include <hip/hip_bf16.h>

#define NUM_CLASSES 19
#define KLD_MARGIN 3.0f
#define DEC 0.01f
#define BOT_EPS 1e-4f
#define TOP_EPS 1.0f

#define LOW 64
#define HI 512
#define TILE 16
#define HALO 18
#define HPOS (HALO * HALO)   // 324
#define PSTRIDE 20           // padded class stride (80B = 16B aligned rows)

typedef float v2f __attribute__((ext_vector_type(2)));
typedef float v8f __attribute__((ext_vector_type(8)));

__device__ __constant__ int c_dy[8] = {-1, -1, -1, 0, 0, 1, 1, 1};
__device__ __constant__ int c_dx[8] = {-1, 0, 1, -1, 1, -1, 0, 1};

// Wave32 sum via V_WMMA_F32_16X16X4_F32 with B = ones.
// A layout (16x4 f32): lanes 0-15 hold M=lane, K={0,1}; lanes 16-31 hold M=lane-16, K={2,3}.
// A = {v, 0} per lane  =>  rowsum[m] = v[m] + v[m+16].
// D layout: lanes<16 hold rows 0-7 in VGPRs 0-7; lanes>=16 hold rows 8-15.
// => lane 0's D-sum = lanes {0-7,16-23}; lane 16's D-sum = lanes {8-15,24-31}.
__device__ inline float wmma_half_wave_sum(float v) {
    v2f a; a.x = v; a.y = 0.0f;
    v2f b; b.x = 1.0f; b.y = 1.0f;
    v8f c = {};
    v8f d = __builtin_amdgcn_wmma_f32_16x16x4_f32(
        false, a, false, b, (short)0, c, false, false);
    return d[0] + d[1] + d[2] + d[3] + d[4] + d[5] + d[6] + d[7];
}

__global__ void aaf_init_kernel(void* ws) {
    double* sums = (double*)ws;
    sums[0] = 0.0;   // edge sum
    sums[1] = 0.0;   // not-edge sum
    unsigned int* cnts = (unsigned int*)(sums + 2);
    cnts[0] = 0u;    // edge count
    cnts[1] = 0u;    // not-edge count
}

__global__ void __launch_bounds__(256)
aaf_main_kernel(const float* __restrict__ preds,
                const int* __restrict__ targets,
                const float* __restrict__ w_edge,
                const float* __restrict__ w_not_edge,
                void* ws) {
    __shared__ float s_prob[HPOS * PSTRIDE];      // halo softmax probs (0 if OOB)
    __shared__ int   s_lab[HPOS];                 // halo labels (-1 if OOB)
    __shared__ float s_we[NUM_CLASSES];
    __shared__ float s_wne[NUM_CLASSES];

    const int tid = threadIdx.x;
    const int bx = blockIdx.x, by = blockIdx.y, n = blockIdx.z;

    // per-class edge/non-edge weights: softmax over last dim (3), take [0]
    if (tid < NUM_CLASSES) {
        const float* r = w_edge + tid * 3;
        float m = fmaxf(r[0], fmaxf(r[1], r[2]));
        float e0 = __expf(r[0] - m), e1 = __expf(r[1] - m), e2 = __expf(r[2] - m);
        s_we[tid] = e0 / (e0 + e1 + e2);
        const float* q = w_not_edge + tid * 3;
        m = fmaxf(q[0], fmaxf(q[1], q[2]));
        e0 = __expf(q[0] - m); e1 = __expf(q[1] - m); e2 = __expf(q[2] - m);
        s_wne[tid] = e0 / (e0 + e1 + e2);
    }

    // Phase 1: fill 18x18 halo with bilinear(align_corners) + softmax probs and labels
    const float* P = preds + (size_t)n * NUM_CLASSES * LOW * LOW;
    const float scl = (float)(LOW - 1) / (float)(HI - 1);  // 63/511
    for (int p = tid; p < HPOS; p += 256) {
        int ly = p / HALO, lx = p - ly * HALO;
        int gy = by * TILE + ly - 1;
        int gx = bx * TILE + lx - 1;
        if (gy >= 0 && gy < HI && gx >= 0 && gx < HI) {
            s_lab[p] = targets[((size_t)n * HI + gy) * HI + gx];
            float sy = gy * scl, sx = gx * scl;
            int y0 = (int)sy, x0 = (int)sx;
            int y1 = min(y0 + 1, LOW - 1), x1 = min(x0 + 1, LOW - 1);
            float wy = sy - (float)y0, wx = sx - (float)x0;
            float l[NUM_CLASSES];
            float mx = -3.4e38f;
            #pragma unroll
            for (int c = 0; c < NUM_CLASSES; ++c) {
                const float* base = P + c * (LOW * LOW);
                float v00 = base[y0 * LOW + x0], v10 = base[y1 * LOW + x0];
                float v01 = base[y0 * LOW + x1], v11 = base[y1 * LOW + x1];
                float row0 = v00 * (1.0f - wy) + v10 * wy;
                float row1 = v01 * (1.0f - wy) + v11 * wy;
                float lv = row0 * (1.0f - wx) + row1 * wx;
                l[c] = lv;
                mx = fmaxf(mx, lv);
            }
            float s = 0.0f;
            #pragma unroll
            for (int c = 0; c < NUM_CLASSES; ++c) { l[c] = __expf(l[c] - mx); s += l[c]; }
            float inv = 1.0f / s;
            #pragma unroll
            for (int c = 0; c < NUM_CLASSES; ++c) s_prob[p * PSTRIDE + c] = l[c] * inv;
        } else {
            s_lab[p] = -1;                       // OOB sentinel (pad: ignore=true, ohl=255)
            #pragma unroll
            for (int c = 0; c < NUM_CLASSES; ++c) s_prob[p * PSTRIDE + c] = 0.0f; // prob pad 0
        }
    }
    __syncthreads();

    // Phase 2: per-pixel KL + masks (branchless inner loop)
    const int ty = tid / TILE, tx = tid - ty * TILE;
    const int hc = (ty + 1) * HALO + (tx + 1);
    const int lab = s_lab[hc];                   // in-bounds, >= 0
    const bool ignc = (lab > NUM_CLASSES - 1);
    const float wem  = ignc ? 0.0f : s_we[lab];
    const float wnem = ignc ? 0.0f : s_wne[lab];

    float lpc[NUM_CLASSES], lnpc[NUM_CLASSES];
    #pragma unroll
    for (int c = 0; c < NUM_CLASSES; ++c) {
        float pr = s_prob[hc * PSTRIDE + c];
        float pc = fminf(fmaxf(pr, BOT_EPS), TOP_EPS);
        float np = fminf(fmaxf(1.0f - pr, BOT_EPS), TOP_EPS);
        lpc[c] = __logf(pc);
        lnpc[c] = __logf(np);
    }

    float e_sum = 0.0f, ne_sum = 0.0f, e_cnt = 0.0f, ne_cnt = 0.0f;

    if (!ignc) {
        #pragma unroll
        for (int k = 0; k < 8; ++k) {
            const int dy = c_dy[k], dx = c_dx[k];
            // bidirectional ignore mask uses the REVERSED offset neighbor (pad=ignored)
            const int hb = (ty + 1 - dy) * HALO + (tx + 1 - dx);
            const int lb = s_lab[hb];
            const bool mask = (lb >= 0) && (lb <= NUM_CLASSES - 1);
            if (!mask) continue;   // nearly wave-uniform (image border only)
            // edge test + probs use the FORWARD offset neighbor
            const int hf = (ty + 1 + dy) * HALO + (tx + 1 + dx);
            const int lf = s_lab[hf];
            const bool nb_valid = (lf >= 0) && (lf <= NUM_CLASSES - 1);

            // edge_c = !nb_valid || ((c==lab) != (c==lf)):
            //   nb_valid && lf==lab -> no edge classes
            //   nb_valid && lf!=lab -> exactly classes {lab, lf} are edges
            //   !nb_valid           -> all 19 classes are edges
            float skld = 0.0f;      // sum of kld over all classes
            float smargin = 0.0f;   // sum of max(0, 3-kld) over all classes
            float kl_lab = 0.0f, kl_lf = 0.0f;
            #pragma unroll
            for (int c = 0; c < NUM_CLASSES; ++c) {
                float pr = s_prob[hf * PSTRIDE + c];            // 0 if OOB (zero pad)
                float ppc  = fminf(fmaxf(pr, BOT_EPS), TOP_EPS);
                float nppc = fminf(fmaxf(1.0f - pr, BOT_EPS), TOP_EPS);
                float kld = ppc * (__logf(ppc) - lpc[c]) + nppc * (__logf(nppc) - lnpc[c]);
                skld += kld;
                smargin += fmaxf(0.0f, KLD_MARGIN - kld);
                kl_lab = (c == lab) ? kld : kl_lab;             // v_cndmask
                kl_lf  = (c == lf)  ? kld : kl_lf;
            }
            const bool same = (lf == lab);
            float ecnt_k = nb_valid ? (same ? 0.0f : 2.0f) : 19.0f;
            float e_add  = nb_valid
                ? (same ? 0.0f
                        : fmaxf(0.0f, KLD_MARGIN - kl_lab) + fmaxf(0.0f, KLD_MARGIN - kl_lf))
                : smargin;
            float ne_add = nb_valid ? (same ? skld : skld - kl_lab - kl_lf) : 0.0f;

            e_sum  += wem * e_add;
            ne_sum += wnem * ne_add;
            e_cnt  += ecnt_k;
            ne_cnt += 19.0f - ecnt_k;
        }
    }

    // Wave32 reduction on the matrix unit (no early exits -> EXEC all ones here)
    float r_es = wmma_half_wave_sum(e_sum);
    float r_ns = wmma_half_wave_sum(ne_sum);
    float r_ec = wmma_half_wave_sum(e_cnt);
    float r_nc = wmma_half_wave_sum(ne_cnt);

    double* sums = (double*)ws;
    unsigned int* cnts = (unsigned int*)(sums + 2);
    const int lane = tid & 31;
    if (lane == 0 || lane == 16) {   // each holds half the wave's total
        atomicAdd(&sums[0], (double)r_es);
        atomicAdd(&sums[1], (double)r_ns);
        atomicAdd(&cnts[0], (unsigned int)(r_ec + 0.5f));
        atomicAdd(&cnts[1], (unsigned int)(r_nc + 0.5f));
    }
}

__global__ void aaf_finalize_kernel(const void* ws, float* out) {
    const double* sums = (const double*)ws;
    const unsigned int* cnts = (const unsigned int*)(sums + 2);
    double ec = (double)max(cnts[0], 1u);
    double nc = (double)max(cnts[1], 1u);
    out[0] = (float)((sums[0] / ec) * (double)DEC + (sums[1] / nc) * (double)DEC);
}

extern "C" void kernel_launch(void* const* d_in, const int* in_sizes, int n_in,
                              void* d_out, int out_size, void* d_ws, size_t ws_size,
                              hipStream_t stream) {
    const float* preds      = (const float*)d_in[0];   // (2,19,64,64)
    const int*   targets    = (const int*)d_in[1];     // (2,512,512)
    const float* w_edge     = (const float*)d_in[2];   // (1,1,1,19,1,3)
    const float* w_not_edge = (const float*)d_in[3];   // (1,1,1,19,1,3)
    float* out = (float*)d_out;

    aaf_init_kernel<<<1, 1, 0, stream>>>(d_ws);

    dim3 grid(HI / TILE, HI / TILE, 2);   // 32 x 32 x 2
    dim3 block(256);
    aaf_main_kernel<<<grid, block, 0, stream>>>(preds, targets, w_edge, w_not_edge, d_ws);

    aaf_finalize_kernel<<<1, 1, 0, stream>>>(d_ws, out);
}